// GARNOLDI_34162169872619
// MI455X (gfx1250) — compile-verified
//
#include <hip/hip_runtime.h>
#include <stdint.h>

// GPR/Arnoldi polynomial propagation for MI455X (gfx1250).
//   hidden = sum_k coeffs[k] * A_hat^k x,  A_hat = D^-1/2 (A+I) D^-1/2
// N=50k nodes, C=128 ch, E=1.6M edges, K=10 hops.
//
// Roofline: 4.2 GFLOP total; x/xhat/edge arrays (~97MB) are L2-resident
// (192MB L2), so the binding resource is the L2 atomic/scatter path
// (~205M f32 atomics per hop). Hot kernel: one wave32 per edge, lane ->
// 4 channels (32 lanes x float4 == 128 ch exactly), per-wave double-
// buffered edge tiles streamed through the CDNA5 async global->LDS copy
// engine (ASYNCcnt), no workgroup barriers. No WMMA: uniform-random graph
// => expected 0.16 edges per 16x16 block, tensor-pipe SpMM would be >99%
// zero work.

#define C_DIM   128
#define WTILE   128      // edges per wave-tile (32 lanes x 4 edges)
#define EBLK    256      // 8 waves per block
#define NWAVES  (EBLK / 32)

// ---------------- setup kernels ----------------

__global__ __launch_bounds__(256) void garnoldi_fill_deg(float* deg, int n) {
  int i = blockIdx.x * blockDim.x + threadIdx.x;
  if (i < n) deg[i] = 1.0f;                    // self-loop contributes 1
}

__global__ __launch_bounds__(256) void garnoldi_deg(const int* __restrict__ dst,
                                                    float* __restrict__ deg, int E) {
  int i = blockIdx.x * blockDim.x + threadIdx.x;
  if (i < E) atomicAdd(&deg[dst[i]], 1.0f);
}

__global__ __launch_bounds__(256) void garnoldi_dinv(const float* __restrict__ deg,
                                                     float* __restrict__ dinv, int n) {
  int i = blockIdx.x * blockDim.x + threadIdx.x;
  if (i < n) {
    float d = deg[i];
    dinv[i] = (d > 0.0f) ? (1.0f / sqrtf(d)) : 0.0f;
  }
}

__global__ __launch_bounds__(256) void garnoldi_edgew(const int* __restrict__ src,
                                                      const int* __restrict__ dst,
                                                      const float* __restrict__ dinv,
                                                      float* __restrict__ w, int E) {
  int i = blockIdx.x * blockDim.x + threadIdx.x;
  if (i < E) w[i] = dinv[src[i]] * dinv[dst[i]];
}

// xa = x ; out = coeffs[0] * x     (vectorized over float4)
__global__ __launch_bounds__(256) void garnoldi_init(const float* __restrict__ x,
                                                     const float* __restrict__ coeffs,
                                                     float* __restrict__ xa,
                                                     float* __restrict__ out, int total4) {
  int i = blockIdx.x * blockDim.x + threadIdx.x;
  if (i < total4) {
    float4 v = reinterpret_cast<const float4*>(x)[i];
    reinterpret_cast<float4*>(xa)[i] = v;
    float c0 = coeffs[0];
    float4 o; o.x = v.x * c0; o.y = v.y * c0; o.z = v.z * c0; o.w = v.w * c0;
    reinterpret_cast<float4*>(out)[i] = o;
  }
}

// xout = dinv^2 * xin   (self-loop term; initializes xout for the edge atomics)
__global__ __launch_bounds__(256) void garnoldi_self(const float* __restrict__ xin,
                                                     const float* __restrict__ dinv,
                                                     float* __restrict__ xout, int total4) {
  int i = blockIdx.x * blockDim.x + threadIdx.x;
  if (i < total4) {
    int node = i / (C_DIM / 4);
    float dv = dinv[node];
    float s  = dv * dv;
    float4 v = reinterpret_cast<const float4*>(xin)[i];
    float4 o; o.x = v.x * s; o.y = v.y * s; o.z = v.z * s; o.w = v.w * s;
    reinterpret_cast<float4*>(xout)[i] = o;
  }
}

// Fused hop epilogue: out += coeffs[k] * xn  AND  xnext = dinv^2 * xn
// (one read of xn feeds both the accumulator update and the next hop's
// self-loop-initialized buffer). writeNext=0 on the final hop.
__global__ __launch_bounds__(256) void garnoldi_fuse(const float* __restrict__ xn,
                                                     const float* __restrict__ dinv,
                                                     const float* __restrict__ coeffs, int k,
                                                     float* __restrict__ out,
                                                     float* __restrict__ xnext,
                                                     int total4, int writeNext) {
  int i = blockIdx.x * blockDim.x + threadIdx.x;
  if (i < total4) {
    float g  = coeffs[k];
    float4 v = reinterpret_cast<const float4*>(xn)[i];
    float4 o = reinterpret_cast<const float4*>(out)[i];
    o.x += g * v.x; o.y += g * v.y; o.z += g * v.z; o.w += g * v.w;
    reinterpret_cast<float4*>(out)[i] = o;
    if (writeNext) {
      int node = i / (C_DIM / 4);
      float dv = dinv[node];
      float s  = dv * dv;
      float4 p; p.x = v.x * s; p.y = v.y * s; p.z = v.z * s; p.w = v.w * s;
      reinterpret_cast<float4*>(xnext)[i] = p;
    }
  }
}

// ---------------- hot kernel: pipelined edge scatter ----------------
// Per-wave private LDS slices, double-buffered, filled by async global->LDS
// b128 copies. Exactly 3 async instructions per wave per tile => static
// ASYNCcnt waits (wait<=3 while the next tile is in flight).

__device__ __forceinline__ void garnoldi_prefetch(const int* __restrict__ src,
                                                  const int* __restrict__ dst,
                                                  const float* __restrict__ w,
                                                  int tile_base, int E, int lane,
                                                  int* lsrc, int* ldst, float* lw) {
  // lane L copies the 16B chunk for edges [4L, 4L+4) of this tile.
  int c = tile_base + lane * 4;
  if (c > E - 4) c = E - 4;          // clamp: always issue, garbage patched later
  if (c < 0) c = 0;
  uint32_t a_src = (uint32_t)(uintptr_t)(lsrc + lane * 4);
  uint32_t a_dst = (uint32_t)(uintptr_t)(ldst + lane * 4);
  uint32_t a_w   = (uint32_t)(uintptr_t)(lw   + lane * 4);
  uint64_t g_src = (uint64_t)(uintptr_t)(src + c);
  uint64_t g_dst = (uint64_t)(uintptr_t)(dst + c);
  uint64_t g_w   = (uint64_t)(uintptr_t)(w   + c);
  asm volatile("global_load_async_to_lds_b128 %0, %1, off"
               :: "v"(a_src), "v"(g_src) : "memory");
  asm volatile("global_load_async_to_lds_b128 %0, %1, off"
               :: "v"(a_dst), "v"(g_dst) : "memory");
  asm volatile("global_load_async_to_lds_b128 %0, %1, off"
               :: "v"(a_w),   "v"(g_w)   : "memory");
}

__global__ __launch_bounds__(EBLK) void garnoldi_edge(const int* __restrict__ src,
                                                      const int* __restrict__ dst,
                                                      const float* __restrict__ w,
                                                      const float* __restrict__ xin,
                                                      float* __restrict__ xout,
                                                      int E, int ntiles) {
  __shared__ int   s_src[2][NWAVES][WTILE];
  __shared__ int   s_dst[2][NWAVES][WTILE];
  __shared__ float s_w  [2][NWAVES][WTILE];

  const int tid    = threadIdx.x;
  const int wave   = tid >> 5;
  const int lane   = tid & 31;
  const int gwave  = blockIdx.x * NWAVES + wave;   // global wave id
  const int stride = gridDim.x * NWAVES;           // wave-tile stride
  const int ch     = lane * 4;                     // this lane's 4 channels

  int t = gwave;
  if (t < ntiles)
    garnoldi_prefetch(src, dst, w, t * WTILE, E, lane,
                      s_src[0][wave], s_dst[0][wave], s_w[0][wave]);

  int buf = 0;
  for (; t < ntiles; t += stride, buf ^= 1) {
    const int tnext = t + stride;
    if (tnext < ntiles) {
      garnoldi_prefetch(src, dst, w, tnext * WTILE, E, lane,
                        s_src[buf ^ 1][wave], s_dst[buf ^ 1][wave], s_w[buf ^ 1][wave]);
      asm volatile("s_wait_asynccnt 3" ::: "memory");  // tile t landed, t+1 in flight
    } else {
      asm volatile("s_wait_asynccnt 0" ::: "memory");  // drain
    }

    const int base = t * WTILE;
    const int n    = min(WTILE, E - base);
    if (n == WTILE) {
      // fast path: full tile, unrolled so several b128 gathers stay in
      // flight (LOADcnt) above the fire-and-forget atomics (STOREcnt).
#pragma unroll 4
      for (int e = 0; e < WTILE; ++e) {
        const int   si = s_src[buf][wave][e];
        const int   di = s_dst[buf][wave][e];
        const float wt = s_w  [buf][wave][e];
        const float4 v = *reinterpret_cast<const float4*>(xin + (size_t)si * C_DIM + ch);
        float* o = xout + (size_t)di * C_DIM + ch;
        atomicAdd(o + 0, v.x * wt);
        atomicAdd(o + 1, v.y * wt);
        atomicAdd(o + 2, v.z * wt);
        atomicAdd(o + 3, v.w * wt);
      }
    } else {
      // ragged tail (never taken when E % WTILE == 0): patch LDS from
      // global, then process the valid prefix.
      for (int j = lane; j < n; j += 32) {
        s_src[buf][wave][j] = src[base + j];
        s_dst[buf][wave][j] = dst[base + j];
        s_w  [buf][wave][j] = w[base + j];
      }
      for (int e = 0; e < n; ++e) {
        const int   si = s_src[buf][wave][e];
        const int   di = s_dst[buf][wave][e];
        const float wt = s_w  [buf][wave][e];
        const float4 v = *reinterpret_cast<const float4*>(xin + (size_t)si * C_DIM + ch);
        float* o = xout + (size_t)di * C_DIM + ch;
        atomicAdd(o + 0, v.x * wt);
        atomicAdd(o + 1, v.y * wt);
        atomicAdd(o + 2, v.z * wt);
        atomicAdd(o + 3, v.w * wt);
      }
    }
  }
}

// ---------------- host orchestration ----------------

extern "C" void kernel_launch(void* const* d_in, const int* in_sizes, int n_in,
                              void* d_out, int out_size, void* d_ws, size_t ws_size,
                              hipStream_t stream) {
  const float* x      = (const float*)d_in[0];
  const float* coeffs = (const float*)d_in[1];
  const int*   ei     = (const int*)d_in[2];

  const int NC  = in_sizes[0];          // N * 128
  const int Kp1 = in_sizes[1];          // K + 1 = 11
  const int E   = in_sizes[2] / 2;      // 1.6M
  const int N   = NC / C_DIM;           // 50000
  const int T4  = NC / 4;               // float4 count

  const int* src = ei;
  const int* dst = ei + E;
  float* out = (float*)d_out;

  // Workspace carve-up (256B aligned): deg | dinv | w | xa | xb  (~58 MB)
  char* ws = (char*)d_ws;
  size_t o = 0;
  auto carve = [&](size_t bytes) {
    void* p = (void*)(ws + o);
    o = (o + bytes + 255) & ~(size_t)255;
    return p;
  };
  float* deg  = (float*)carve((size_t)N * 4);
  float* dinv = (float*)carve((size_t)N * 4);
  float* w    = (float*)carve((size_t)E * 4);
  float* xa   = (float*)carve((size_t)NC * 4);
  float* xb   = (float*)carve((size_t)NC * 4);
  (void)ws_size;

  const int gN  = (N  + 255) / 256;
  const int gE  = (E  + 255) / 256;
  const int gT4 = (T4 + 255) / 256;

  const int ntiles = (E + WTILE - 1) / WTILE;            // 12500 wave-tiles
  int gEdge = (ntiles + NWAVES * 2 - 1) / (NWAVES * 2);  // ~2 tiles per wave
  if (gEdge < 1) gEdge = 1;

  // GCN normalization
  garnoldi_fill_deg<<<gN, 256, 0, stream>>>(deg, N);
  garnoldi_deg<<<gE, 256, 0, stream>>>(dst, deg, E);
  garnoldi_dinv<<<gN, 256, 0, stream>>>(deg, dinv, N);
  garnoldi_edgew<<<gE, 256, 0, stream>>>(src, dst, dinv, w, E);

  // hidden0 = coeffs[0] * x ; xa = x
  garnoldi_init<<<gT4, 256, 0, stream>>>(x, coeffs, xa, out, T4);

  // First hop's self-loop init, then K pipelined hops with fused epilogue.
  //   invariant at loop top: xcur = propagated state, xnxt = dinv^2 * xcur
  float* xcur = xa;
  float* xnxt = xb;
  garnoldi_self<<<gT4, 256, 0, stream>>>(xcur, dinv, xnxt, T4);
  for (int k = 1; k < Kp1; ++k) {
    garnoldi_edge<<<gEdge, EBLK, 0, stream>>>(src, dst, w, xcur, xnxt, E, ntiles);
    // out += coeffs[k]*xnxt; and (unless last hop) xcur := dinv^2*xnxt for hop k+1
    garnoldi_fuse<<<gT4, 256, 0, stream>>>(xnxt, dinv, coeffs, k, out, xcur,
                                           T4, (k + 1 < Kp1) ? 1 : 0);
    float* t = xcur; xcur = xnxt; xnxt = t;              // ping-pong
  }
}